// FocalLoss_76124000354538
// MI455X (gfx1250) — compile-verified
//
#include <hip/hip_runtime.h>
#include <hip/hip_bf16.h>

// ---------------------------------------------------------------------------
// FocalLoss: mean over B of  -(1 - Pt)^2 * log2(Pt),  Pt = input[i, target[i]]
// B = 65536, C = 1024 (derived from in_sizes at launch).
//
// Latency-bound scattered gather -> one thread per row, 2048 wave32s in
// flight. Deterministic 2-pass reduction. Wave-level reduction implemented
// with V_WMMA_F32_16X16X4_F32 (ones-matrix row-sum trick) to use the CDNA5
// matrix pipe.
// ---------------------------------------------------------------------------

typedef __attribute__((ext_vector_type(2))) float v2f;
typedef __attribute__((ext_vector_type(8))) float v8f;

// Full 32-lane sum of `v`, result broadcast to every lane.
// A (16x4 f32): lane L<16 holds A[L][0] (=v), A[L][1]=0;
//               lane L>=16 holds A[L-16][2] (=v), A[L-16][3]=0.
// B = ones(4x16)  =>  D[m][n] = v(lane m) + v(lane m+16)  for all n.
// C/D layout: lane n<16, VGPR r = D[r][n]; lane n>=16, VGPR r = D[r+8][n-16].
// So sum of the 8 D VGPRs per lane = sum over that lane's 8 M-rows; one
// xor-16 shuffle folds the two halves -> total wave sum.
__device__ __forceinline__ float wave_sum_wmma(float v) {
    v2f a;
    a.x = v;
    a.y = 0.0f;
    v2f b;
    b.x = 1.0f;
    b.y = 1.0f;
    v8f c = {};
    // (neg_a, A, neg_b, B, c_mod, C, reuse_a, reuse_b)
    v8f d = __builtin_amdgcn_wmma_f32_16x16x4_f32(false, a, false, b,
                                                  (short)0, c, false, false);
    float s = d[0] + d[1] + d[2] + d[3] + d[4] + d[5] + d[6] + d[7];
    s += __shfl_xor(s, 16, 32);
    return s;
}

__global__ __launch_bounds__(256) void focal_partial_kernel(
    const float* __restrict__ inp, const int* __restrict__ tgt,
    float* __restrict__ partial, int B, int C) {
    const int i = blockIdx.x * 256 + threadIdx.x;

    float loss = 0.0f;
    if (i < B) {
        const int t = tgt[i];                       // coalesced
        const float pt = inp[(size_t)i * (size_t)C + (size_t)t];  // scattered gather
        const float om = 1.0f - pt;
        // v_log_f32 is log base 2 == log(pt) * LOG2E in the reference
        loss = -om * om * __log2f(pt);
    }

    // EXEC all-ones here (guard closed above; tail lanes contribute 0).
    const float wsum = wave_sum_wmma(loss);

    __shared__ float smem[8];
    const int lane = threadIdx.x & 31;
    const int wave = threadIdx.x >> 5;
    if (lane == 0) smem[wave] = wsum;
    __syncthreads();
    if (threadIdx.x == 0) {
        float s = 0.0f;
#pragma unroll
        for (int w = 0; w < 8; ++w) s += smem[w];
        partial[blockIdx.x] = s;
    }
}

__global__ __launch_bounds__(256) void focal_finish_kernel(
    const float* __restrict__ partial, float* __restrict__ out, int n,
    float inv_b) {
    // Single block of 256 threads; grid-stride accumulate then reduce.
    float v = 0.0f;
    for (int idx = threadIdx.x; idx < n; idx += 256) v += partial[idx];

    const float wsum = wave_sum_wmma(v);

    __shared__ float smem[8];
    const int lane = threadIdx.x & 31;
    const int wave = threadIdx.x >> 5;
    if (lane == 0) smem[wave] = wsum;
    __syncthreads();
    if (threadIdx.x == 0) {
        float s = 0.0f;
#pragma unroll
        for (int w = 0; w < 8; ++w) s += smem[w];
        out[0] = s * inv_b;
    }
}

extern "C" void kernel_launch(void* const* d_in, const int* in_sizes, int n_in,
                              void* d_out, int out_size, void* d_ws,
                              size_t ws_size, hipStream_t stream) {
    const float* inp = (const float*)d_in[0];
    const int* tgt = (const int*)d_in[1];
    float* out = (float*)d_out;
    float* partial = (float*)d_ws;

    const int B = in_sizes[1];             // 65536
    const int C = in_sizes[0] / in_sizes[1];  // 1024

    const int blocks = (B + 255) / 256;    // 256 blocks -> 256 partials in d_ws

    focal_partial_kernel<<<blocks, 256, 0, stream>>>(inp, tgt, partial, B, C);
    focal_finish_kernel<<<1, 256, 0, stream>>>(partial, out, blocks,
                                               1.0f / (float)B);
}